// GCN_17575006175346
// MI455X (gfx1250) — compile-verified
//
#include <hip/hip_runtime.h>
#include <math.h>
#include <stdint.h>

// ---------------- problem constants (from reference) ----------------
#define N_NODES 50000
#define N_EDGES 800000
#define IN_C 64
#define HID_C 64
#define OUT_C 40

typedef __attribute__((ext_vector_type(2)))  float    v2f;
typedef __attribute__((ext_vector_type(8)))  float    v8f;

// ---------------- native f32 atomic add (no CAS expansion risk) ----------------
// Scatter-add is the workload bottleneck (~51M atomics/layer); force the native
// VMEM atomic: GLOBAL_ATOMIC_ADD_F32, no-return form (tracked with STOREcnt,
// drained by the implicit wait-idle at s_endpgm).
__device__ __forceinline__ void atomic_add_f32(float* p, float v) {
  asm volatile("global_atomic_add_f32 %0, %1, off" :: "v"(p), "v"(v) : "memory");
}

__device__ __forceinline__ void wait_async0() {
#if __has_builtin(__builtin_amdgcn_s_wait_asynccnt)
  __builtin_amdgcn_s_wait_asynccnt(0);
#else
  asm volatile("s_wait_asynccnt 0" ::: "memory");
#endif
}

// ---------------- helpers ----------------
__global__ void k_zero(float* __restrict__ p, int n) {
  int i = blockIdx.x * blockDim.x + threadIdx.x;
  if (i < n) p[i] = 0.0f;
}

__global__ void k_degree(const int* __restrict__ dst, float* __restrict__ deg, int nE) {
  int e = blockIdx.x * blockDim.x + threadIdx.x;
  if (e < nE) atomic_add_f32(&deg[dst[e]], 1.0f);
}

__global__ void k_rsqrt(float* __restrict__ deg, int n) {
  int i = blockIdx.x * blockDim.x + threadIdx.x;
  if (i < n) deg[i] = rsqrtf(deg[i] + 1.0f);  // -> dis = (deg+1)^-1/2 (self-loop)
}

// ---------------- GEMM: H[n, stride 64] = X[n,64] @ W[64,C], via f32 WMMA ----------------
// 256 threads = 8 waves; wave w owns row-tile (blockIdx.x*8 + w) of 16 rows and sweeps
// C in 16-col tiles. W staged into LDS with async loads (ASYNCcnt path); padded stride
// 65 keeps the two half-wave K-offset DS reads bank-conflict free.
__global__ __launch_bounds__(256) void k_gemm_wmma(const float* __restrict__ X,
                                                   const float* __restrict__ W,
                                                   float* __restrict__ H,
                                                   int nRows, int C) {
  __shared__ float Wl[64 * 65];
  const int Cp = (C + 15) & ~15;
  const int tid = threadIdx.x;
  for (int i = tid; i < 64 * 64; i += 256) {
    int k = i >> 6, c = i & 63;
    if (c < C) {
      const float* gp = W + k * C + c;
      unsigned loff = (unsigned)(uintptr_t)&Wl[k * 65 + c];  // low 32 bits of LDS-aperture addr = LDS offset
      asm volatile("global_load_async_to_lds_b32 %0, %1, off" :: "v"(loff), "v"(gp) : "memory");
    } else {
      Wl[k * 65 + c] = 0.0f;  // zero-pad cols C..63 (layer 3: C=40)
    }
  }
  wait_async0();
  __syncthreads();

  const int lane = tid & 31;
  const int wv   = tid >> 5;
  const int rt   = blockIdx.x * 8 + wv;
  if (rt * 16 >= nRows) return;  // whole wave exits uniformly; EXEC stays all-1s for WMMA

  const int mrow = rt * 16 + (lane & 15);
  const float* __restrict__ xrow = X + mrow * 64;

  // V_WMMA_F32_16X16X4_F32, 16 steps of K=4 (exact f32).
  // A 16x4 layout: lane L holds A[L%16][kb], A[L%16][kb+1], kb = (L/16)*2
  const int kb = (lane >> 4) * 2;
  v2f afrag[16];
#pragma unroll
  for (int t = 0; t < 16; ++t) {
    afrag[t].x = xrow[t * 4 + kb];
    afrag[t].y = xrow[t * 4 + kb + 1];
  }
  for (int ct = 0; ct < Cp / 16; ++ct) {
    const int col = ct * 16 + (lane & 15);
    v8f acc = {0.f, 0.f, 0.f, 0.f, 0.f, 0.f, 0.f, 0.f};
#pragma unroll
    for (int t = 0; t < 16; ++t) {
      v2f b;
      b.x = Wl[(t * 4 + kb) * 65 + col];
      b.y = Wl[(t * 4 + kb + 1) * 65 + col];
      acc = __builtin_amdgcn_wmma_f32_16x16x4_f32(false, afrag[t], false, b,
                                                  (short)0, acc, false, false);
    }
    const int mb = rt * 16 + (lane >> 4) * 8;
    const int n  = ct * 16 + (lane & 15);
    if (n < C) {
      float* hp = H + mb * 64 + n;
#pragma unroll
      for (int j = 0; j < 8; ++j) hp[j * 64] = acc[j];  // D VGPR j: M = mb+j, N = n
    }
  }
}

// ---------------- edge scatter: AGG[dst] += H[src] * dis[src]*dis[dst] ----------------
// one thread per (edge, 4-channel group); coalesced float4 gather, 4 native f32 atomics.
template <int NG>
__global__ __launch_bounds__(256) void k_scatter(const float* __restrict__ H,
                                                 const int* __restrict__ src,
                                                 const int* __restrict__ dst,
                                                 const float* __restrict__ dis,
                                                 float* __restrict__ AGG,
                                                 int nE) {
  int t = blockIdx.x * blockDim.x + threadIdx.x;
  int e = t / NG;          // NG=16 -> shifts; NG=10 -> magic multiply
  int g = t - e * NG;
  if (e >= nE) return;
  const int s = src[e], d = dst[e];
  const float norm = dis[s] * dis[d];
  const float4 hv = *(const float4*)(H + s * 64 + g * 4);
  float* ap = AGG + d * 64 + g * 4;
  atomic_add_f32(ap + 0, hv.x * norm);
  atomic_add_f32(ap + 1, hv.y * norm);
  atomic_add_f32(ap + 2, hv.z * norm);
  atomic_add_f32(ap + 3, hv.w * norm);
}

// ---------------- finalize (layers 1,2): Y = relu(AGG + H*dis^2 + b); AGG := 0 ----------------
__global__ void k_finalize_relu(const float* __restrict__ H, float* __restrict__ AGG,
                                const float* __restrict__ dis, const float* __restrict__ b,
                                float* __restrict__ Y, int nNodes) {
  int i = blockIdx.x * blockDim.x + threadIdx.x;
  if (i >= nNodes * 64) return;
  const int node = i >> 6, c = i & 63;
  const float ds = dis[node];
  const float v = AGG[i] + H[i] * ds * ds + b[c];
  Y[i] = v > 0.0f ? v : 0.0f;
  AGG[i] = 0.0f;  // pre-zero for next layer's scatter
}

// ---------------- finalize (layer 3): log_softmax over 40 channels ----------------
__global__ void k_logsoftmax(const float* __restrict__ H, const float* __restrict__ AGG,
                             const float* __restrict__ dis, const float* __restrict__ b,
                             float* __restrict__ OUT, int nNodes) {
  int node = blockIdx.x * blockDim.x + threadIdx.x;
  if (node >= nNodes) return;
  const float ds = dis[node];
  const float ds2 = ds * ds;
  float v[OUT_C];
  float m = -1e30f;
#pragma unroll
  for (int c = 0; c < OUT_C; ++c) {
    float t = AGG[node * 64 + c] + H[node * 64 + c] * ds2 + b[c];
    v[c] = t;
    m = fmaxf(m, t);
  }
  float s = 0.0f;
#pragma unroll
  for (int c = 0; c < OUT_C; ++c) s += __expf(v[c] - m);
  const float ls = __logf(s);
#pragma unroll
  for (int c = 0; c < OUT_C; ++c) OUT[node * OUT_C + c] = v[c] - m - ls;
}

// ---------------- launcher ----------------
extern "C" void kernel_launch(void* const* d_in, const int* in_sizes, int n_in,
                              void* d_out, int out_size, void* d_ws, size_t ws_size,
                              hipStream_t stream) {
  const float* x   = (const float*)d_in[0];
  const int*   ei  = (const int*)d_in[1];  // [2, E] row-major: src row then dst row
  const float* W1  = (const float*)d_in[2];
  const float* b1  = (const float*)d_in[3];
  const float* W2  = (const float*)d_in[4];
  const float* b2  = (const float*)d_in[5];
  const float* W3  = (const float*)d_in[6];
  const float* b3  = (const float*)d_in[7];
  float* out = (float*)d_out;

  const int* src = ei;
  const int* dst = ei + N_EDGES;

  float* ws  = (float*)d_ws;
  float* deg = ws;                       // N_NODES floats (becomes dis after k_rsqrt)
  float* H   = ws + 50048;               // N_NODES*64 (stride 64, layer-3 uses cols 0..39)
  float* AGG = H + (size_t)N_NODES * 64; // N_NODES*64
  float* Y   = AGG + (size_t)N_NODES * 64;

  const int B = 256;
  const int gNodeC  = (N_NODES * 64 + B - 1) / B;
  const int gNodes  = (N_NODES + B - 1) / B;
  const int gEdges  = (N_EDGES + B - 1) / B;
  const int gGemm   = (N_NODES / 16 + 7) / 8;      // 3125 row tiles / 8 waves
  const int gScat64 = (N_EDGES * 16 + B - 1) / B;  // 64ch -> 16 float4 groups
  const int gScat40 = (N_EDGES * 10 + B - 1) / B;  // 40ch -> 10 float4 groups

  // fresh state each launch (ws is not re-poisoned between replays)
  k_zero<<<gNodes, B, 0, stream>>>(deg, N_NODES);
  k_zero<<<gNodeC, B, 0, stream>>>(AGG, N_NODES * 64);

  k_degree<<<gEdges, B, 0, stream>>>(dst, deg, N_EDGES);
  k_rsqrt<<<gNodes, B, 0, stream>>>(deg, N_NODES);

  // layer 1
  k_gemm_wmma<<<gGemm, B, 0, stream>>>(x, W1, H, N_NODES, HID_C);
  k_scatter<16><<<gScat64, B, 0, stream>>>(H, src, dst, deg, AGG, N_EDGES);
  k_finalize_relu<<<gNodeC, B, 0, stream>>>(H, AGG, deg, b1, Y, N_NODES);

  // layer 2
  k_gemm_wmma<<<gGemm, B, 0, stream>>>(Y, W2, H, N_NODES, HID_C);
  k_scatter<16><<<gScat64, B, 0, stream>>>(H, src, dst, deg, AGG, N_EDGES);
  k_finalize_relu<<<gNodeC, B, 0, stream>>>(H, AGG, deg, b2, Y, N_NODES);

  // layer 3 + log_softmax
  k_gemm_wmma<<<gGemm, B, 0, stream>>>(Y, W3, H, N_NODES, OUT_C);
  k_scatter<10><<<gScat40, B, 0, stream>>>(H, src, dst, deg, AGG, N_EDGES);
  k_logsoftmax<<<gNodes, B, 0, stream>>>(H, AGG, deg, b3, out, N_NODES);
}